// SpectralPredictor_62474594288002
// MI455X (gfx1250) — compile-verified
//
#include <hip/hip_runtime.h>
#include <cstdint>

// Problem constants
#define ZB   200
#define YB   128
#define XB   128
#define ZP_S 240   // sA z-padding: [0..15]=0, [16..215]=image, [216..239]=0
#define ZP_P 224   // pS band-padding: [0..199]=psum, [200..223]=don't-care (finite)

// -------- CDNA5 async global->LDS staging (guarded) --------------------------
#if defined(__HIP_DEVICE_COMPILE__) && defined(__gfx1250__) && \
    __has_builtin(__builtin_amdgcn_global_load_async_to_lds_b128)
  #define HAVE_ASYNC 1
#else
  #define HAVE_ASYNC 0
#endif

#if HAVE_ASYNC
  #define GAS1 __attribute__((address_space(1)))
  #define LAS3 __attribute__((address_space(3)))
  typedef int v4i __attribute__((ext_vector_type(4)));
  __device__ __forceinline__ void async_copy16(const float* g, float* l) {
    __builtin_amdgcn_global_load_async_to_lds_b128(
        (GAS1 v4i*)(g), (LAS3 v4i*)(l), 0, 0);
  }
  __device__ __forceinline__ void wait_async() {
  #if __has_builtin(__builtin_amdgcn_s_wait_asynccnt)
    __builtin_amdgcn_s_wait_asynccnt(0);
  #else
    asm volatile("s_wait_asynccnt 0" ::: "memory");
  #endif
  }
#else
  __device__ __forceinline__ void async_copy16(const float* g, float* l) {
    *(float4*)l = *(const float4*)g;   // global_load_b128 + ds_store_b128
  }
  __device__ __forceinline__ void wait_async() {}
#endif

// -------- CDNA5 WMMA availability -------------------------------------------
#if defined(__HIP_DEVICE_COMPILE__) && defined(__gfx1250__) && \
    __has_builtin(__builtin_amdgcn_wmma_f32_16x16x4_f32)
  #define HAVE_WMMA 1
#else
  #define HAVE_WMMA 0
#endif

typedef float v2f __attribute__((ext_vector_type(2)));
typedef float v8f __attribute__((ext_vector_type(8)));

// -------- weight update: w' = clip(w + e*d*2^{-4-max(0,V-6)}, -2, 2) --------
__device__ __forceinline__ float updf(float w, float d, float e) {
  int E = (__float_as_int(d) & 0x7fffffff) >> 23;  // biased exponent of |d|
  int t = (E > 133) ? E : 133;                     // 133 = 127 + V_MAX
  float scale = __int_as_float((256 - t) << 23);   // 2^(-4 - max(0, V-6))
  float wn = fmaf(e * d, scale, w);
  return fminf(fmaxf(wn, -2.0f), 2.0f);
}

// -------- pack kernel: z-transpose with zero pads ---------------------------
// sA[(y*XB+x)*ZP_S + (b+16)] = image[b][y][x]
__global__ __launch_bounds__(256) void SpectralPredictor_pack(
    const float* __restrict__ img, float* __restrict__ sA) {
  int tid = blockIdx.x * 256 + threadIdx.x;
  if (tid >= ZB * YB * XB) return;
  int b = tid >> 14;            // band
  int r = tid & 16383;          // y*XB + x  (lane-contiguous in x -> coalesced)
  sA[(size_t)r * ZP_S + b + 16] = img[(size_t)b * 16384 + r];
  if (b < 16) sA[(size_t)r * ZP_S + b] = 0.0f;
  if (b < 24) sA[(size_t)r * ZP_S + 216 + b] = 0.0f;
}

// -------- psum kernel: banded matmul via V_WMMA_F32_16X16X4_F32 --------------
// pS[(y*XB+x)*ZP_P + b] = sum_{k=1..14, b-k>=0} image[b-k][y][x]
// psum = M * img where M[b][b'] = 1 iff 1 <= b-b' <= 14 (and b' in range).
// One wave per 16x16 (band-tile x pixel-tile) output tile; the 29-wide band
// means K spans two 16-tiles around the diagonal -> 8 chained 16x16x4 WMMAs.
__global__ __launch_bounds__(32) void SpectralPredictor_psum(
    const float* __restrict__ img, float* __restrict__ pS) {
  const int bt   = blockIdx.y * 16;     // band tile base (14 tiles -> b 0..223)
  const int rt   = blockIdx.x * 16;     // pixel tile base (1024 tiles)
  const int lane = threadIdx.x;
  const int half = lane >> 4;           // K half-split across lane halves
  const int nm   = lane & 15;           // M for A, N for B/D (per f32 layouts)
#if HAVE_WMMA
  v8f acc = {};                         // C = 0
  #pragma unroll
  for (int c = 0; c < 8; ++c) {         // K chunks of 4 covering b' in [bt-16, bt+15]
    v2f av, bv;
    #pragma unroll
    for (int j = 0; j < 2; ++j) {
      int bp   = (bt - 16 + c * 4) + half * 2 + j;   // global b' for this (lane, vgpr)
      int diff = (bt + nm) - bp;                     // b - b'
      av[j] = (bp >= 0 && diff >= 1 && diff <= 14) ? 1.0f : 0.0f;
      int bpc = bp < 0 ? 0 : (bp > ZB - 1 ? ZB - 1 : bp);  // clamp address; av==0 there
      bv[j] = img[(size_t)bpc * 16384 + rt + nm];          // coalesced per half-wave
    }
    acc = __builtin_amdgcn_wmma_f32_16x16x4_f32(
        false, av, false, bv, (short)0, acc, false, false);
  }
  // D layout: lane -> N=nm, vgpr m8 -> M = m8 + half*8
  #pragma unroll
  for (int m8 = 0; m8 < 8; ++m8) {
    int m = m8 + half * 8;
    pS[(size_t)(rt + nm) * ZP_P + (bt + m)] = acc[m8];
  }
#else
  // scalar fallback with identical launch geometry
  for (int idx = lane; idx < 256; idx += 32) {
    int m = idx >> 4, n = idx & 15;
    int b = bt + m;
    float ps = 0.0f;
    if (b < ZB) {
      for (int k = 1; k <= 14; ++k) {
        int bp = b - k;
        if (bp >= 0) ps += img[(size_t)bp * 16384 + rt + n];
      }
    }
    pS[(size_t)(rt + n) * ZP_P + b] = ps;
  }
#endif
}

// -------- scan kernel: 1 wave per workgroup, 1 band per lane ----------------
__global__ __launch_bounds__(32) void SpectralPredictor_scan(
    const float* __restrict__ sA, const float* __restrict__ pS,
    float* __restrict__ out) {
  __shared__ __align__(16) float buf[3][XB * 32];   // 48 KB: s-row ring buffer

  const int lane = threadIdx.x;        // 0..31
  const int zb   = blockIdx.x * 32;    // band base of this wave
  const int z    = zb + lane;          // this lane's band (may be >= 200)
  const bool zok = (z < ZB);

  // per-lane constants
  const float cntf = (float)((z < 14) ? z : 14);     // # valid spec p=1..14
  const float mp0  = (z >= 15) ? 1.0f : 0.0f;        // validity of d_spec[0]
  const float du   = (z >= 1) ? 1.0f : 0.0f;         // do_update

  // async row staging: lane covers x = xg*4 + (lane>>3), z-quad (lane&7)*4
  const int xs = lane >> 3;
  const int zo = (lane & 7) * 4;
  auto stage = [&](int yy) {
    float* dstb = buf[yy % 3];
    const float* g = sA + (size_t)yy * (XB * ZP_S) + (zb + 16);
    #pragma unroll
    for (int xg = 0; xg < 32; ++xg) {
      int x = xg * 4 + xs;
      async_copy16(g + (size_t)x * ZP_S + zo, dstb + x * 32 + zo);
    }
  };

  float w0 = 0.f, w1 = 0.f, w2 = 0.f, w3 = 0.f, wsp = 0.f;
  const size_t zc = zok ? (size_t)z : 0;             // safe store base

  stage(0);
  for (int y = 0; y < YB; ++y) {
    wait_async();                                    // buf[y%3] ready
    if (y + 1 < YB) stage(y + 1);                    // overlap with this row's scan

    const float* cur = buf[y % 3];
    const float* prv = buf[(y + 2) % 3];             // row y-1 (valid for y>0)
    float* orow = out + zc * 16384 + (size_t)y * XB;
    const int pixbase = y * XB;

    // spatial registers (zero-padded boundaries)
    float W_ = 0.f, NW_ = 0.f, N_, NE_;
    if (y > 0) { N_ = prv[lane]; NE_ = prv[32 + lane]; }
    else       { N_ = 0.f;       NE_ = 0.f; }

    auto substep = [&](int x) -> float {
      float s = cur[x * 32 + lane];                  // ds_load, bank=lane
      float sigma;
      if (y == 0)        sigma = (x > 0) ? 4.0f * W_ : 0.0f;
      else if (x == 0)   sigma = 2.0f * (N_ + NE_);
      else if (x == 127) sigma = 2.0f * (W_ + NW_);
      else               sigma = ((NW_ + NE_) + W_) + N_;

      size_t pix = (size_t)(pixbase + x);
      float psum = pS[pix * ZP_P + (size_t)z];           // coalesced b32
      float pv15 = sA[pix * ZP_S + (size_t)(z + 1)];     // coalesced b32

      float dN   = fmaf(4.0f, N_,   -sigma);
      float dW   = fmaf(4.0f, W_,   -sigma);
      float dNW  = fmaf(4.0f, NW_,  -sigma);
      float dp0  = mp0 * fmaf(4.0f, pv15, -sigma);
      float Dsum = fmaf(4.0f, psum, -(cntf * sigma));

      float acc  = fmaf(w0, dN, w1 * dW) + fmaf(w2, dNW, w3 * dp0)
                 + fmaf(wsp, Dsum, 0.25f * sigma);
      float pred = fminf(fmaxf(acc, -32768.0f), 32767.0f);
      float e    = (s - pred) * du;

      w0 = updf(w0, dN,  e);
      w1 = updf(w1, dW,  e);
      w2 = updf(w2, dNW, e);
      w3 = updf(w3, dp0, e);
      wsp = fminf(fmaxf(fmaf(e, 0.00625f, wsp), -2.0f), 2.0f);

      // rotate spatial registers for x+1
      NW_ = N_; N_ = NE_; W_ = s;
      NE_ = (y > 0 && x + 2 < XB) ? prv[(x + 2) * 32 + lane] : 0.0f;
      return pred;
    };

    for (int x4 = 0; x4 < XB; x4 += 4) {
      float p0 = substep(x4 + 0);
      float p1 = substep(x4 + 1);
      float p2 = substep(x4 + 2);
      float p3 = substep(x4 + 3);
      if (zok) {
        float4 v = make_float4(p0, p1, p2, p3);
        *(float4*)(orow + x4) = v;                   // 16B-aligned b128 store
      }
    }
  }
}

extern "C" void kernel_launch(void* const* d_in, const int* in_sizes, int n_in,
                              void* d_out, int out_size, void* d_ws, size_t ws_size,
                              hipStream_t stream) {
  (void)in_sizes; (void)n_in; (void)out_size; (void)ws_size;
  const float* img = (const float*)d_in[0];
  float* out = (float*)d_out;
  // workspace: sA (YB*XB*ZP_S floats) | pS (YB*XB*ZP_P floats)  ~30.4 MB
  float* sA = (float*)d_ws;
  float* pS = sA + (size_t)YB * XB * ZP_S;

  int n = ZB * YB * XB;
  SpectralPredictor_pack<<<(n + 255) / 256, 256, 0, stream>>>(img, sA);
  dim3 pg(YB * XB / 16, ZP_P / 16);   // 1024 x 14 tiles, one wave each
  SpectralPredictor_psum<<<pg, 32, 0, stream>>>(img, pS);
  SpectralPredictor_scan<<<(ZB + 31) / 32, 32, 0, stream>>>(sA, pS, out);
}